// GHMC_51591147159894
// MI455X (gfx1250) — compile-verified
//
#include <hip/hip_runtime.h>
#include <math.h>

// GHM-C loss for MI455X (gfx1250, wave32).
// Pass 1: global 10-bin histogram of g=|sigmoid(pred)-target| over valid samples
//         (per-wave LDS sub-histograms -> integer global atomics, deterministic).
// Pass 2: recompute g/bin + stable BCE, weight by per-bin coefficient, block
//         partial sums (wave reduction via V_WMMA_F32_16X16X4_F32 with B=ones).
// Pass 3: deterministic final reduce of block partials, divide by tot.

#define BINS 10
#define TPB 256
#define NWAVES (TPB / 32)
#define GRID_HIST 2048
#define GRID_LOSS 2048

typedef __attribute__((ext_vector_type(2))) float v2f;
typedef __attribute__((ext_vector_type(8))) float v8f;

// Full wave32 sum using one f32 WMMA.
// A (16x4, f32): lanes 0-15 hold K=0 (vgpr0), K=1 (vgpr1); lanes 16-31 hold K=2,3.
// With a[0]=x, a[1]=0 and B=ones: D[m][n] = x[lane m] + x[lane m+16].
// D layout: lanes 0-15 vgpr v = row M=v; lanes 16-31 vgpr v = row M=v+8.
// Sum of the 8 D vgprs per lane + shfl_xor(16) => total over all 32 lanes.
// Requires EXEC all-ones: call only with every thread of the wave active.
__device__ __forceinline__ float wave_sum_wmma(float x) {
  v2f a;
  a[0] = x;
  a[1] = 0.0f;
  v2f b;
  b[0] = 1.0f;
  b[1] = 1.0f;
  v8f c = {};
  v8f d = __builtin_amdgcn_wmma_f32_16x16x4_f32(
      /*neg_a=*/false, a, /*neg_b=*/false, b,
      /*c_mod=*/(short)0, c, /*reuse_a=*/false, /*reuse_b=*/false);
  float s = d[0] + d[1] + d[2] + d[3] + d[4] + d[5] + d[6] + d[7];
  s += __shfl_xor(s, 16, 32);
  return s;
}

__device__ __forceinline__ int bin_of(float p, float t) {
  float sg = 1.0f / (1.0f + expf(-p));
  float g = fabsf(sg - t);
  int b = (int)(g * 10.0f);      // floor(g*BINS), g >= 0
  b = b > (BINS - 1) ? (BINS - 1) : b;
  return b < 0 ? 0 : b;
}

__device__ __forceinline__ float bce_of(float p, float t) {
  // max(p,0) - p*t + log1p(exp(-|p|))
  return fmaxf(p, 0.0f) - p * t + log1pf(expf(-fabsf(p)));
}

__global__ void ghmc_zero(unsigned int* __restrict__ counts) {
  if (threadIdx.x < BINS) counts[threadIdx.x] = 0u;
}

__global__ void ghmc_hist(const float* __restrict__ pred,
                          const float* __restrict__ tgt,
                          const float* __restrict__ lw,
                          unsigned int* __restrict__ counts, long long n) {
  __shared__ unsigned int h[NWAVES * 16];  // padded per-wave sub-histograms
  for (int i = threadIdx.x; i < NWAVES * 16; i += blockDim.x) h[i] = 0u;
  __syncthreads();
  unsigned int* wh = &h[(threadIdx.x >> 5) * 16];

  const long long nv = n >> 2;
  const long long idx = (long long)blockIdx.x * blockDim.x + threadIdx.x;
  const long long stride = (long long)gridDim.x * blockDim.x;
  const float4* p4 = (const float4*)pred;
  const float4* t4 = (const float4*)tgt;
  const float4* w4 = (const float4*)lw;

  for (long long i = idx; i < nv; i += stride) {
    __builtin_prefetch(p4 + i + stride, 0, 1);
    __builtin_prefetch(t4 + i + stride, 0, 1);
    __builtin_prefetch(w4 + i + stride, 0, 1);
    float4 p = p4[i], t = t4[i], w = w4[i];
    if (w.x > 0.0f) atomicAdd(&wh[bin_of(p.x, t.x)], 1u);
    if (w.y > 0.0f) atomicAdd(&wh[bin_of(p.y, t.y)], 1u);
    if (w.z > 0.0f) atomicAdd(&wh[bin_of(p.z, t.z)], 1u);
    if (w.w > 0.0f) atomicAdd(&wh[bin_of(p.w, t.w)], 1u);
  }
  // scalar tail (empty when n % 4 == 0)
  for (long long i = (nv << 2) + idx; i < n; i += stride) {
    if (lw[i] > 0.0f) atomicAdd(&wh[bin_of(pred[i], tgt[i])], 1u);
  }

  __syncthreads();
  for (int b = threadIdx.x; b < BINS; b += blockDim.x) {
    unsigned int s = 0;
    for (int w = 0; w < NWAVES; ++w) s += h[w * 16 + b];
    atomicAdd(&counts[b], s);  // integer atomics: deterministic
  }
}

__global__ void ghmc_loss(const float* __restrict__ pred,
                          const float* __restrict__ tgt,
                          const float* __restrict__ lw,
                          const unsigned int* __restrict__ counts,
                          float* __restrict__ partials, long long n) {
  __shared__ float scoef[16];
  __shared__ float wsum[NWAVES];
  if (threadIdx.x == 0) {
    unsigned int c[BINS];
    unsigned int tot = 0;
    int nb = 0;
    for (int b = 0; b < BINS; ++b) {
      c[b] = counts[b];
      tot += c[b];                 // every valid sample lands in exactly one bin
      if (c[b] != 0u) ++nb;
    }
    float totf = fmaxf((float)tot, 1.0f);
    float nf = fmaxf((float)nb, 1.0f);
    for (int b = 0; b < BINS; ++b)
      scoef[b] = (c[b] != 0u) ? (totf / (float)c[b]) / nf : 0.0f;
  }
  __syncthreads();

  float lsum = 0.0f;
  const long long nv = n >> 2;
  const long long idx = (long long)blockIdx.x * blockDim.x + threadIdx.x;
  const long long stride = (long long)gridDim.x * blockDim.x;
  const float4* p4 = (const float4*)pred;
  const float4* t4 = (const float4*)tgt;
  const float4* w4 = (const float4*)lw;

  for (long long i = idx; i < nv; i += stride) {
    __builtin_prefetch(p4 + i + stride, 0, 1);
    __builtin_prefetch(t4 + i + stride, 0, 1);
    __builtin_prefetch(w4 + i + stride, 0, 1);
    float4 p = p4[i], t = t4[i], w = w4[i];
    if (w.x > 0.0f) lsum += scoef[bin_of(p.x, t.x)] * bce_of(p.x, t.x);
    if (w.y > 0.0f) lsum += scoef[bin_of(p.y, t.y)] * bce_of(p.y, t.y);
    if (w.z > 0.0f) lsum += scoef[bin_of(p.z, t.z)] * bce_of(p.z, t.z);
    if (w.w > 0.0f) lsum += scoef[bin_of(p.w, t.w)] * bce_of(p.w, t.w);
  }
  for (long long i = (nv << 2) + idx; i < n; i += stride) {
    if (lw[i] > 0.0f) lsum += scoef[bin_of(pred[i], tgt[i])] * bce_of(pred[i], tgt[i]);
  }

  // All threads active here: WMMA wave reduction is legal (EXEC all ones).
  float s = wave_sum_wmma(lsum);
  if ((threadIdx.x & 31) == 0) wsum[threadIdx.x >> 5] = s;
  __syncthreads();
  if (threadIdx.x == 0) {
    float t = 0.0f;
    for (int w = 0; w < NWAVES; ++w) t += wsum[w];  // fixed order: deterministic
    partials[blockIdx.x] = t;
  }
}

__global__ void ghmc_final(const float* __restrict__ partials, int np,
                           const unsigned int* __restrict__ counts,
                           float* __restrict__ out) {
  __shared__ float wsum[NWAVES];
  float lsum = 0.0f;
  for (int i = threadIdx.x; i < np; i += blockDim.x) lsum += partials[i];
  float s = wave_sum_wmma(lsum);
  if ((threadIdx.x & 31) == 0) wsum[threadIdx.x >> 5] = s;
  __syncthreads();
  if (threadIdx.x == 0) {
    float t = 0.0f;
    for (int w = 0; w < NWAVES; ++w) t += wsum[w];
    unsigned int tot = 0;
    for (int b = 0; b < BINS; ++b) tot += counts[b];
    float totf = fmaxf((float)tot, 1.0f);
    out[0] = (t / totf) * 1.0f;  // LOSS_WEIGHT = 1.0
  }
}

extern "C" void kernel_launch(void* const* d_in, const int* in_sizes, int n_in,
                              void* d_out, int out_size, void* d_ws, size_t ws_size,
                              hipStream_t stream) {
  const float* pred = (const float*)d_in[0];
  const float* tgt = (const float*)d_in[1];
  const float* lw = (const float*)d_in[2];
  const long long n = (long long)in_sizes[0];

  unsigned int* counts = (unsigned int*)d_ws;                    // 10 uints
  float* partials = (float*)((char*)d_ws + 256);                 // GRID_LOSS floats

  ghmc_zero<<<1, 32, 0, stream>>>(counts);
  ghmc_hist<<<GRID_HIST, TPB, 0, stream>>>(pred, tgt, lw, counts, n);
  ghmc_loss<<<GRID_LOSS, TPB, 0, stream>>>(pred, tgt, lw, counts, partials, n);
  ghmc_final<<<1, TPB, 0, stream>>>(partials, GRID_LOSS, counts, (float*)d_out);
}